// JPEGCoding_70695161692177
// MI455X (gfx1250) — compile-verified
//
#include <hip/hip_runtime.h>

typedef float v2f __attribute__((ext_vector_type(2)));
typedef float v8f __attribute__((ext_vector_type(8)));
typedef unsigned int u32x4 __attribute__((ext_vector_type(4)));
typedef unsigned int u32x8 __attribute__((ext_vector_type(8)));

#define IMG_B 8
#define IMG_H 1024
#define IMG_W 1024

// JPEG base quantization tables (luma, chroma) as float (no v_cvt in hot loop)
__constant__ float cQTf[2][64] = {
  {16,11,10,16,24,40,51,61,
   12,12,14,19,26,58,60,55,
   14,13,16,24,40,57,69,56,
   14,17,22,29,51,87,80,62,
   18,22,37,56,68,109,103,77,
   24,35,55,64,81,104,113,92,
   49,64,78,87,103,121,120,101,
   72,92,95,98,112,100,103,99},
  {17,18,24,47,99,99,99,99,
   18,21,26,66,99,99,99,99,
   24,26,56,99,99,99,99,99,
   47,66,99,99,99,99,99,99,
   99,99,99,99,99,99,99,99,
   99,99,99,99,99,99,99,99,
   99,99,99,99,99,99,99,99,
   99,99,99,99,99,99,99,99}};

// Block-diagonal 16x16 DCT matrix: diag(D8, D8), D8[k][n] = 0.5*cos((2n+1)k*pi/16) (*1/sqrt2 for k=0)
__device__ __forceinline__ float dblk(int r, int c) {
  if ((r >> 3) != (c >> 3)) return 0.0f;
  int k = r & 7, n = c & 7;
  float v = 0.5f * __cosf((float)((2 * n + 1) * k) * 0.19634954084936207f);
  if (k == 0) v *= 0.70710678118654752f;
  return v;
}

// ---------------- Kernel 1: RGB -> YCbCr + 2x2 chroma subsample ----------------
__global__ __launch_bounds__(256) void k_rgb2ycc(const float* __restrict__ img,
                                                 float* __restrict__ yP,
                                                 float* __restrict__ cbP,
                                                 float* __restrict__ crP) {
  const int HW = IMG_H * IMG_W;
  const int H2 = IMG_H / 2, W2 = IMG_W / 2;
  int t = blockIdx.x * 256 + threadIdx.x;          // one 2x2 quad per thread
  int b = t / (H2 * W2);
  int rem = t - b * (H2 * W2);
  int r2 = rem / W2;
  int c2 = rem - r2 * W2;
  int r0 = r2 * 2, c0 = c2 * 2;

  const float* base = img + (size_t)b * 3 * HW + (size_t)r0 * IMG_W + c0;
  float2 R0 = *(const float2*)(base);
  float2 R1 = *(const float2*)(base + IMG_W);
  float2 G0 = *(const float2*)(base + HW);
  float2 G1 = *(const float2*)(base + HW + IMG_W);
  float2 B0 = *(const float2*)(base + 2 * HW);
  float2 B1 = *(const float2*)(base + 2 * HW + IMG_W);

  float2 Y0, Y1;
  Y0.x = 0.299f * R0.x + 0.587f * G0.x + 0.114f * B0.x;
  Y0.y = 0.299f * R0.y + 0.587f * G0.y + 0.114f * B0.y;
  Y1.x = 0.299f * R1.x + 0.587f * G1.x + 0.114f * B1.x;
  Y1.y = 0.299f * R1.y + 0.587f * G1.y + 0.114f * B1.y;

  // 2x2 mean commutes with the linear color transform
  float rs = (R0.x + R0.y + R1.x + R1.y) * 0.25f;
  float gs = (G0.x + G0.y + G1.x + G1.y) * 0.25f;
  float bs = (B0.x + B0.y + B1.x + B1.y) * 0.25f;
  float cb = -0.168736f * rs - 0.331264f * gs + 0.5f * bs + 128.0f;
  float cr = 0.5f * rs - 0.418688f * gs - 0.081312f * bs + 128.0f;

  float* yb = yP + (size_t)b * HW + (size_t)r0 * IMG_W + c0;
  *(float2*)(yb) = Y0;
  *(float2*)(yb + IMG_W) = Y1;
  size_t ci = (size_t)b * H2 * W2 + (size_t)r2 * W2 + c2;
  cbP[ci] = cb;
  crP[ci] = cr;
}

// ---------------- Kernel 2: per-channel 8x8 DCT + diff-round quant + IDCT ------
// One wave per 16x16 tile (= 4 JPEG blocks) using V_WMMA_F32_16X16X4_F32.
// Tile is fetched global->LDS by the Tensor Data Mover (TDM), with TDM LDS
// padding (1 DWORD per 16) producing the stride-17 bank-conflict-free layout.
__device__ __forceinline__ v8f gemm16(const float a[4][2], const float b[4][2]) {
  v8f acc = {0.f, 0.f, 0.f, 0.f, 0.f, 0.f, 0.f, 0.f};
#pragma unroll
  for (int c = 0; c < 4; ++c) {
    v2f av = {a[c][0], a[c][1]};
    v2f bv = {b[c][0], b[c][1]};
    acc = __builtin_amdgcn_wmma_f32_16x16x4_f32(false, av, false, bv,
                                                (short)0, acc, false, false);
  }
  return acc;
}

#define LSTRIDE 17  // padded LDS row stride (floats): gcd(17,64)=1 -> conflict-free

__global__ __launch_bounds__(256) void k_dct(float* __restrict__ plane,
                                             const float* __restrict__ strength,
                                             int planeH, int planeW, int isChroma) {
  __shared__ float lds[8][16 * LSTRIDE];
  const int wave = threadIdx.x >> 5;
  const int lid = threadIdx.x & 31;
  const int half = lid >> 4;
  const int l16 = lid & 15;

  const int tilesX = planeW >> 4;
  const int tpi = tilesX * (planeH >> 4);
  const int tile = blockIdx.x * 8 + wave;
  const int b = tile / tpi;
  int rem = tile - b * tpi;
  const int ty = rem / tilesX;
  const int tx = rem - ty * tilesX;

  float* pl = plane + (size_t)b * planeH * planeW + (size_t)(ty * 16) * planeW + tx * 16;
  float* T = lds[wave];

  // ---- TDM descriptor (wave-uniform, built in SGPRs) ----
  unsigned long long ga = (unsigned long long)(uintptr_t)pl;  // tile global byte addr
  unsigned int ga_lo = (unsigned int)__builtin_amdgcn_readfirstlane((int)(ga & 0xffffffffu));
  unsigned int ga_hi = (unsigned int)__builtin_amdgcn_readfirstlane((int)(ga >> 32));
  unsigned int laddr = (unsigned int)__builtin_amdgcn_readfirstlane((int)(uintptr_t)T);
  unsigned int pw = (unsigned int)planeW, ph = (unsigned int)planeH;

  u32x4 g0;
  g0.x = 1u;                                        // count=1, user descriptor
  g0.y = laddr;                                     // lds_addr (bytes)
  g0.z = ga_lo;                                     // global_addr[31:0]
  g0.w = (ga_hi & 0x01ffffffu) | 0x80000000u;       // global_addr[56:32] | type=2

  u32x8 g1;
  g1[0] = (2u << 16)        // data_size = 4B
        | (1u << 20)        // pad_enable
        | (3u << 22);       // pad_interval = 16 DWORDs (pad_amount=0 -> 1 DWORD)
  g1[1] = pw << 16;                                 // tensor_dim0[15:0]
  g1[2] = (ph & 0xffffu) << 16;                     // tensor_dim0[31:16]=0 | tensor_dim1[15:0]
  g1[3] = 16u << 16;                                // tensor_dim1[31:16]=0 | tile_dim0=16
  g1[4] = 16u;                                      // tile_dim1=16, tile_dim2=0
  g1[5] = pw;                                       // tensor_dim0_stride[31:0]
  g1[6] = 0u;                                       // stride[47:32]=0 | dim1_stride lo
  g1[7] = 0u;                                       // dim1_stride hi

  u32x4 gz = {0u, 0u, 0u, 0u};                      // groups 2/3: 2-D tensor, unused

  asm volatile("tensor_load_to_lds %0, %1, %2, %3"
               :
               : "s"(g0), "s"(g1), "s"(gz), "s"(gz)
               : "memory");

  // quant scale for this image (overlaps TDM)
  float q = strength[b];
  float s = (q < 50.0f) ? (5000.0f / q) : (200.0f - 2.0f * q);

  // Constant DCT fragments (computed, not loaded):
  //  fD [c][j] = Dblk[l16][k]  : A-frag of Dblk   AND B-frag of Dblk^T
  //  fDT[c][j] = Dblk[k][l16]  : A-frag of Dblk^T AND B-frag of Dblk
  float fD[4][2], fDT[4][2];
#pragma unroll
  for (int c = 0; c < 4; ++c)
#pragma unroll
    for (int j = 0; j < 2; ++j) {
      int k = 4 * c + j + 2 * half;
      fD[c][j] = dblk(l16, k);
      fDT[c][j] = dblk(k, l16);
    }

  __builtin_amdgcn_s_wait_tensorcnt(0);             // tile resident in LDS

  // Stage 1: T1 = Dblk * X     (B-frag of raw X from LDS; no -128 needed, see DC shift)
  float fB[4][2], fA[4][2];
#pragma unroll
  for (int c = 0; c < 4; ++c)
#pragma unroll
    for (int j = 0; j < 2; ++j)
      fB[c][j] = T[(4 * c + j + 2 * half) * LSTRIDE + l16];
  v8f t1 = gemm16(fD, fB);

  // Stage 2: C = T1 * Dblk^T   (A-frag of T1 via LDS roundtrip, B-frag = fD)
#pragma unroll
  for (int v = 0; v < 8; ++v) T[(v + 8 * half) * LSTRIDE + l16] = t1[v];
#pragma unroll
  for (int c = 0; c < 4; ++c)
#pragma unroll
    for (int j = 0; j < 2; ++j)
      fA[c][j] = T[l16 * LSTRIDE + 4 * c + j + 2 * half];
  v8f coef = gemm16(fA, fD);

  // Quantize in C/D layout: lane knows (M,N) -> qt index (M&7, N&7).
  // The -128 pixel shift only moves each block's DC term by 128*8 = 1024
  // (D8 rows k>0 sum to zero), so apply +-1024 at DC instead of +-128 per pixel.
#pragma unroll
  for (int v = 0; v < 8; ++v) {
    int M = v + 8 * half;
    int qi = (M & 7) * 8 + (l16 & 7);
    float base = isChroma ? cQTf[1][qi] : cQTf[0][qi];
    float qt = floorf((base * s + 50.0f) * 0.01f);   // STE floor (forward)
    qt = fminf(fmaxf(qt, 1.0f), 255.0f);
    float dc = (((M & 7) | (l16 & 7)) == 0) ? 1024.0f : 0.0f;
    float x = (coef[v] - dc) * __builtin_amdgcn_rcpf(qt);  // fast v_rcp_f32
    float r = rintf(x);                              // round-half-even == jnp.round
    float e = x - r;
    coef[v] = (r + e * e * e) * qt + dc;             // differentiable round (+DC shift back)
  }

  // Stage 3: T2 = Dblk^T * C   (B-frag of C via LDS roundtrip, A-frag = fDT)
#pragma unroll
  for (int v = 0; v < 8; ++v) T[(v + 8 * half) * LSTRIDE + l16] = coef[v];
#pragma unroll
  for (int c = 0; c < 4; ++c)
#pragma unroll
    for (int j = 0; j < 2; ++j)
      fB[c][j] = T[(4 * c + j + 2 * half) * LSTRIDE + l16];
  v8f t2 = gemm16(fDT, fB);

  // Stage 4: R = T2 * Dblk     (A-frag of T2 via LDS roundtrip, B-frag = fDT)
#pragma unroll
  for (int v = 0; v < 8; ++v) T[(v + 8 * half) * LSTRIDE + l16] = t2[v];
#pragma unroll
  for (int c = 0; c < 4; ++c)
#pragma unroll
    for (int j = 0; j < 2; ++j)
      fA[c][j] = T[l16 * LSTRIDE + 4 * c + j + 2 * half];
  v8f rec = gemm16(fA, fDT);

  // Write reconstructed tile back in place (DC shift already restored the +128)
#pragma unroll
  for (int v = 0; v < 8; ++v)
    pl[(size_t)(v + 8 * half) * planeW + l16] = rec[v];
}

// ---------------- Kernel 3: upsample chroma + YCbCr -> RGB + clip --------------
__device__ __forceinline__ float clip255(float v) {
  return fminf(fmaxf(v, 0.0f), 255.0f);
}

__global__ __launch_bounds__(256) void k_ycc2rgb(const float* __restrict__ yP,
                                                 const float* __restrict__ cbP,
                                                 const float* __restrict__ crP,
                                                 float* __restrict__ out) {
  const int HW = IMG_H * IMG_W;
  const int H2 = IMG_H / 2, W2 = IMG_W / 2;
  int t = blockIdx.x * 256 + threadIdx.x;
  int b = t / (H2 * W2);
  int rem = t - b * (H2 * W2);
  int r2 = rem / W2;
  int c2 = rem - r2 * W2;
  int r0 = r2 * 2, c0 = c2 * 2;

  size_t ci = (size_t)b * H2 * W2 + (size_t)r2 * W2 + c2;
  float cb = cbP[ci] - 128.0f;
  float cr = crP[ci] - 128.0f;

  const float* yb = yP + (size_t)b * HW + (size_t)r0 * IMG_W + c0;
  float2 Y0 = *(const float2*)(yb);
  float2 Y1 = *(const float2*)(yb + IMG_W);

  float kr = 1.402f * cr;
  float kg = -0.344136f * cb - 0.714136f * cr;
  float kb = 1.772f * cb;

  float2 Rv0 = {clip255(Y0.x + kr), clip255(Y0.y + kr)};
  float2 Rv1 = {clip255(Y1.x + kr), clip255(Y1.y + kr)};
  float2 Gv0 = {clip255(Y0.x + kg), clip255(Y0.y + kg)};
  float2 Gv1 = {clip255(Y1.x + kg), clip255(Y1.y + kg)};
  float2 Bv0 = {clip255(Y0.x + kb), clip255(Y0.y + kb)};
  float2 Bv1 = {clip255(Y1.x + kb), clip255(Y1.y + kb)};

  float* ob = out + (size_t)b * 3 * HW + (size_t)r0 * IMG_W + c0;
  *(float2*)(ob) = Rv0;
  *(float2*)(ob + IMG_W) = Rv1;
  *(float2*)(ob + HW) = Gv0;
  *(float2*)(ob + HW + IMG_W) = Gv1;
  *(float2*)(ob + 2 * HW) = Bv0;
  *(float2*)(ob + 2 * HW + IMG_W) = Bv1;
}

// -------------------------------------------------------------------------------
extern "C" void kernel_launch(void* const* d_in, const int* in_sizes, int n_in,
                              void* d_out, int out_size, void* d_ws, size_t ws_size,
                              hipStream_t stream) {
  const float* img = (const float*)d_in[0];       // (8,3,1024,1024) f32
  const float* strength = (const float*)d_in[1];  // (8,) f32
  float* out = (float*)d_out;                     // (8,3,1024,1024) f32
  float* ws = (float*)d_ws;

  const size_t HW = (size_t)IMG_H * IMG_W;
  const size_t H2W2 = (size_t)(IMG_H / 2) * (IMG_W / 2);
  float* yP = ws;                                  // 8*1024*1024
  float* cbP = yP + (size_t)IMG_B * HW;            // 8*512*512
  float* crP = cbP + (size_t)IMG_B * H2W2;         // 8*512*512

  const int quads = IMG_B * (IMG_H / 2) * (IMG_W / 2);      // 2,097,152
  const int yTiles = IMG_B * (IMG_H / 16) * (IMG_W / 16);   // 32768
  const int cTiles = IMG_B * (IMG_H / 32) * (IMG_W / 32);   // 8192

  k_rgb2ycc<<<quads / 256, 256, 0, stream>>>(img, yP, cbP, crP);
  k_dct<<<yTiles / 8, 256, 0, stream>>>(yP, strength, IMG_H, IMG_W, 0);
  k_dct<<<cTiles / 8, 256, 0, stream>>>(cbP, strength, IMG_H / 2, IMG_W / 2, 1);
  k_dct<<<cTiles / 8, 256, 0, stream>>>(crP, strength, IMG_H / 2, IMG_W / 2, 1);
  k_ycc2rgb<<<quads / 256, 256, 0, stream>>>(yP, cbP, crP, out);
}